// DCTDecomposition_24567212933345
// MI455X (gfx1250) — compile-verified
//
#include <hip/hip_runtime.h>

typedef float v2f __attribute__((ext_vector_type(2)));
typedef float v8f __attribute__((ext_vector_type(8)));

// 8x8 JPEG zigzag order indices, row-major (matches reference _zigzag(8)).
__constant__ int g_zz8[64] = {
     0,  1,  5,  6, 14, 15, 27, 28,
     2,  4,  7, 13, 16, 26, 29, 42,
     3,  8, 12, 17, 25, 30, 41, 43,
     9, 11, 18, 24, 31, 40, 44, 53,
    10, 19, 23, 32, 39, 45, 52, 54,
    20, 22, 33, 38, 46, 51, 55, 60,
    21, 34, 37, 47, 50, 56, 59, 61,
    35, 36, 48, 49, 57, 58, 62, 63
};

// DCT-II matrix entry D[r][c], n=8.  sqrt(1/8)=0.353553, sqrt(2/8)=0.5, pi/16.
__device__ __forceinline__ float dctv(int r, int c) {
    if (r == 0) return 0.35355339059327373f;
    return 0.5f * __cosf(0.19634954084936207f * (float)(r * (2 * c + 1)));
}

// Wait for all DS (LDS) ops of this wave to complete; also a compiler memory fence.
__device__ __forceinline__ void lds_fence() {
    asm volatile("s_wait_dscnt 0" ::: "memory");
}

// 16x16x16 fp32 matmul via 4 chained v_wmma_f32_16x16x4_f32 (K-split accumulate).
__device__ __forceinline__ v8f mm16(const v2f a[4], const v2f b[4]) {
    v8f c = {0.f, 0.f, 0.f, 0.f, 0.f, 0.f, 0.f, 0.f};
#pragma unroll
    for (int s = 0; s < 4; ++s) {
        c = __builtin_amdgcn_wmma_f32_16x16x4_f32(
                /*neg_a=*/false, a[s], /*neg_b=*/false, b[s],
                /*c_mod=*/(short)0, c, /*reuse_a=*/false, /*reuse_b=*/false);
    }
    return c;
}

// C/D layout (16x16 f32): element (m,n) -> vgpr m%8, lane (n + 16*(m>=8)).
// Per lane (p = lane&15, h = lane>>4): slot v holds (m = v+8h, n = p).
__device__ __forceinline__ void storeC_lds(float* lds, int h, int p, v8f c) {
#pragma unroll
    for (int v = 0; v < 8; ++v)
        lds[(v + 8 * h) * 16 + p] = c[v];
}

// A layout (16x16 over 4 steps): slot (s,j) holds Mat[m][k], m = lane&15, k = 4s+j+2h.
__device__ __forceinline__ void loadA_lds(const float* lds, int h, int m, v2f a[4]) {
#pragma unroll
    for (int s = 0; s < 4; ++s)
#pragma unroll
        for (int j = 0; j < 2; ++j)
            a[s][j] = lds[m * 16 + (4 * s + j + 2 * h)];
}

// B layout (16x16 over 4 steps): slot (s,j) holds Mat[k][n], n = lane&15, k = 4s+j+2h.
__device__ __forceinline__ void loadB_lds(const float* lds, int h, int n, v2f b[4]) {
#pragma unroll
    for (int s = 0; s < 4; ++s)
#pragma unroll
        for (int j = 0; j < 2; ++j)
            b[s][j] = lds[(4 * s + j + 2 * h) * 16 + n];
}

__global__ __launch_bounds__(256)
void dct_band_wmma_kernel(const float* __restrict__ x,
                          const float* __restrict__ band_scale,
                          float* __restrict__ out,
                          int numWaves) {
    constexpr int W    = 512;
    constexpr int IMG  = 512 * 512;        // elems per (b,c) image
    constexpr int NT   = 48 * 32 * 32;     // total 16x16 tiles = 49152
    constexpr long OUTB = 16L * 3 * 512 * 512; // elems per output band

    __shared__ float ldsArr[8 * 256];      // 1KB tile buffer per wave (8 waves)

    const int lane   = threadIdx.x & 31;
    const int wvBlk  = threadIdx.x >> 5;
    const int wave   = blockIdx.x * (blockDim.x >> 5) + wvBlk;
    float* lds = &ldsArr[wvBlk * 256];

    const int p = lane & 15;   // m (A-lane / C-row-col) or n (B/C col) index
    const int h = lane >> 4;

    // ---- per-wave constants -------------------------------------------------
    // CF: D2 in A-layout == D2^T in B-layout  (value dctv(p&7, k&7) on diag blocks)
    // CI: D2^T in A-layout == D2 in B-layout  (value dctv(k&7, p&7) on diag blocks)
    // ML/MM: low/mid band masks in B-layout (mask of coeff (k, n=p)).
    v2f CF[4], CI[4], ML[4], MM[4];
#pragma unroll
    for (int s = 0; s < 4; ++s) {
#pragma unroll
        for (int j = 0; j < 2; ++j) {
            int k = 4 * s + j + 2 * h;
            bool blk = (p >> 3) == (k >> 3);       // block-diagonal of diag(D,D)
            CF[s][j] = blk ? dctv(p & 7, k & 7) : 0.f;
            CI[s][j] = blk ? dctv(k & 7, p & 7) : 0.f;
            int zi = g_zz8[(k & 7) * 8 + (p & 7)];
            ML[s][j] = (zi < 21) ? 1.f : 0.f;                  // 64//3
            MM[s][j] = (zi >= 21 && zi < 42) ? 1.f : 0.f;      // 2*64//3
        }
    }
    const float s0 = band_scale[0], s1 = band_scale[1], s2 = band_scale[2];

    // ---- tile loop (uniform per wave -> EXEC all-ones for WMMA) -------------
    for (int t = wave; t < NT; t += numWaves) {
        const int img = t >> 10;          // /1024 tiles per image
        const int ti  = t & 1023;
        const int tr  = ti >> 5, tc = ti & 31;
        const long base = (long)img * IMG + (long)tr * (16 * W) + tc * 16;

        // Load X tile in C-layout (each global_load_b32: two contiguous 64B rows)
        const float* xp = x + base;
        v8f Xc;
#pragma unroll
        for (int v = 0; v < 8; ++v)
            Xc[v] = xp[(v + 8 * h) * W + p];

        // X: C-layout -> LDS -> B-layout
        storeC_lds(lds, h, p, Xc);
        lds_fence();
        v2f Xb[4];  loadB_lds(lds, h, p, Xb);

        // Y = D2 @ X
        v8f Y = mm16(CF, Xb);

        // Y -> A-layout; Ct = Y @ D2^T  (full 2-D DCT coefficients)
        storeC_lds(lds, h, p, Y);
        lds_fence();
        v2f Ya[4];  loadA_lds(lds, h, p, Ya);
        v8f Ct = mm16(Ya, CF);

        // Ct -> B-layout; apply band masks elementwise
        storeC_lds(lds, h, p, Ct);
        lds_fence();
        v2f Cb[4];  loadB_lds(lds, h, p, Cb);
        v2f CLb[4], CMb[4];
#pragma unroll
        for (int s = 0; s < 4; ++s)
#pragma unroll
            for (int j = 0; j < 2; ++j) {
                CLb[s][j] = Cb[s][j] * ML[s][j];
                CMb[s][j] = Cb[s][j] * MM[s][j];
            }

        // L = D2^T @ (Ct*Mlow) @ D2
        v8f Pl = mm16(CI, CLb);
        storeC_lds(lds, h, p, Pl);
        lds_fence();
        v2f Pla[4]; loadA_lds(lds, h, p, Pla);
        v8f L = mm16(Pla, CI);

        // M = D2^T @ (Ct*Mmid) @ D2
        v8f Pm = mm16(CI, CMb);
        storeC_lds(lds, h, p, Pm);
        lds_fence();
        v2f Pma[4]; loadA_lds(lds, h, p, Pma);
        v8f M = mm16(Pma, CI);

        // high = X - low_u - mid_u (orthogonality); scale and store all bands.
        float* oL = out + base;
        float* oM = out + OUTB + base;
        float* oH = out + 2 * OUTB + base;
#pragma unroll
        for (int v = 0; v < 8; ++v) {
            const int off = (v + 8 * h) * W + p;
            const float lv = L[v], mv = M[v];
            oL[off] = lv * s0;
            oM[off] = mv * s1;
            oH[off] = (Xc[v] - lv - mv) * s2;
        }
    }
}

extern "C" void kernel_launch(void* const* d_in, const int* in_sizes, int n_in,
                              void* d_out, int out_size, void* d_ws, size_t ws_size,
                              hipStream_t stream) {
    const float* x  = (const float*)d_in[0];   // [16,3,512,512] f32
    const float* bs = (const float*)d_in[1];   // [3] f32
    float* out = (float*)d_out;                // [3][16,3,512,512] f32 concat

    const int blocks  = 1536;                  // 1536 * 8 waves = 12288 waves
    const int threads = 256;
    const int numWaves = blocks * (threads / 32);

    hipLaunchKernelGGL(dct_band_wmma_kernel, dim3(blocks), dim3(threads), 0, stream,
                       x, bs, out, numWaves);
}